// Brain_86638080295078
// MI455X (gfx1250) — compile-verified
//
#include <hip/hip_runtime.h>
#include <hip/hip_bf16.h>
#include <cstdint>

typedef __attribute__((ext_vector_type(2))) float v2f;
typedef __attribute__((ext_vector_type(8))) float v8f;
typedef int v4i __attribute__((vector_size(16)));   // int32x4, matches builtin param

#define AS1 __attribute__((address_space(1)))
#define AS3 __attribute__((address_space(3)))

#define Hd   4096
#define Dd   8196
#define RTOT 16384              // 4*Hd gate rows
#define TK   64                 // K-tile width (fp32 elements)
#define LDSP 68                 // padded LDS row stride: 16B-aligned rows + conflict-free b64 reads
#define NT1  ((Dd + TK - 1) / TK)   // 129 tiles from W_ih (last is 4 cols)
#define NT2  (Hd / TK)              // 64 tiles from W_hh
#define NT   (NT1 + NT2)            // 193

#if defined(__has_builtin)
#  if __has_builtin(__builtin_amdgcn_global_load_async_to_lds_b128)
#    define HAS_ASYNC 1
#  endif
#endif
#ifndef HAS_ASYNC
#  define HAS_ASYNC 0
#endif

__device__ __forceinline__ void wait_async_le8() {
#if HAS_ASYNC
#  if __has_builtin(__builtin_amdgcn_s_wait_asynccnt)
    __builtin_amdgcn_s_wait_asynccnt(8);
#  else
    asm volatile("s_wait_asynccnt 0x8" ::: "memory");
#  endif
#endif
}
__device__ __forceinline__ void wait_async_le0() {
#if HAS_ASYNC
#  if __has_builtin(__builtin_amdgcn_s_wait_asynccnt)
    __builtin_amdgcn_s_wait_asynccnt(0);
#  else
    asm volatile("s_wait_asynccnt 0x0" ::: "memory");
#  endif
#endif
}

struct TileInfo { const float* W; const float* vec; int K; int k0; };

__device__ __forceinline__ TileInfo tile_info(int t, const float* W_ih, const float* x,
                                              const float* W_hh, const float* h0) {
    TileInfo ti;
    if (t < NT1) { ti.W = W_ih; ti.vec = x;  ti.K = Dd; ti.k0 = t * TK; }
    else         { ti.W = W_hh; ti.vec = h0; ti.K = Hd; ti.k0 = (t - NT1) * TK; }
    return ti;
}

// Stage one private 16x64 fp32 tile for this wave: 8 x (32 lanes x 16B) = 4 KB.
// Async path: GLOBAL_LOAD_ASYNC_TO_LDS_B128 (ASYNCcnt); fallback: b128 via VGPRs.
__device__ __forceinline__ void stage_tile(float (*tl)[LDSP], const float* W,
                                           int K, int k0, int m0, int lane) {
    #pragma unroll
    for (int i = 0; i < 8; ++i) {
        const int j    = i * 32 + lane;     // chunk id 0..255
        const int row  = j >> 4;            // 0..15
        const int colf = (j & 15) * 4;      // 0..60
        int kc = k0 + colf;
        if (kc + 3 >= K) kc = 0;            // clamp tail reads (values never consumed)
        const float* src = W + (long long)(m0 + row) * (long long)K + kc;
        float* dst = &tl[row][colf];        // 16B aligned: (row*68 + colf)*4
#if HAS_ASYNC
        __builtin_amdgcn_global_load_async_to_lds_b128(
            (AS1 v4i*)(uintptr_t)src, (AS3 v4i*)(uintptr_t)dst,
            /*offset=*/0, /*cpol=*/0);
#else
        *(float4*)dst = *(const float4*)src;
#endif
    }
}

__device__ __forceinline__ v8f wmma_step(const float (*tl)[LDSP], const float* __restrict__ vec,
                                         int k0, int kk, int mrow, int half, v8f acc) {
    v2f a, b;
    const int c = kk * 4 + 2 * half;
    a.x = tl[mrow][c + 0];                  // A 16x4: lane = M row, K pair per half
    a.y = tl[mrow][c + 1];
    const int kb = k0 + c;
    b.x = vec[kb + 0];                      // B 4x16: broadcast vec[k] to all 16 cols
    b.y = vec[kb + 1];
    return __builtin_amdgcn_wmma_f32_16x16x4_f32(false, a, false, b, (short)0, acc,
                                                 false, false);
}

// ---------------------------------------------------------------------------
// Kernel 1: gates[16384] = W_ih@x + W_hh@h0 + b_ih + b_hh
// 1024 blocks x 8 wave32s; each block owns 16 gate rows; each wave owns a
// private double-buffered LDS tile and an interleaved K-range. Barrier-free
// main loop; one block barrier for the cross-wave reduction.
// ---------------------------------------------------------------------------
__global__ __launch_bounds__(256)
void lstm_gates_wmma(const float* __restrict__ x, const float* __restrict__ h0,
                     const float* __restrict__ W_ih, const float* __restrict__ W_hh,
                     const float* __restrict__ b_ih, const float* __restrict__ b_hh,
                     float* __restrict__ gates)
{
    __shared__ float tile[2][8][16][LDSP];  // 68 KB: per-wave double buffers
    __shared__ float red[8][16];

    const int tid  = threadIdx.x;
    const int lane = tid & 31;
    const int wave = tid >> 5;              // 0..7 (uniform per wave32)
    const int half = lane >> 4;
    const int mrow = lane & 15;
    const int m0   = blockIdx.x * 16;

    v8f acc = {};

    int t = wave, buf = 0;
    if (t < NT) {
        TileInfo ti = tile_info(t, W_ih, x, W_hh, h0);
        stage_tile(tile[0][wave], ti.W, ti.K, ti.k0, m0, lane);
    }
    for (; t < NT; t += 8) {
        const int tn = t + 8;
        if (tn < NT) {                      // prefetch next tile into other buffer
            TileInfo tj = tile_info(tn, W_ih, x, W_hh, h0);
            stage_tile(tile[buf ^ 1][wave], tj.W, tj.K, tj.k0, m0, lane);
            wait_async_le8();               // in-order completion => current tile ready
        } else {
            wait_async_le0();
        }
        TileInfo ti = tile_info(t, W_ih, x, W_hh, h0);
        const int rem   = ti.K - ti.k0;
        const int steps = (rem < TK ? rem : TK) >> 2;
        const float (*tl)[LDSP] = tile[buf][wave];
        if (steps == 16) {
            #pragma unroll
            for (int kk = 0; kk < 16; ++kk)
                acc = wmma_step(tl, ti.vec, ti.k0, kk, mrow, half, acc);
        } else {
            #pragma unroll 1
            for (int kk = 0; kk < steps; ++kk)
                acc = wmma_step(tl, ti.vec, ti.k0, kk, mrow, half, acc);
        }
        buf ^= 1;
    }

    // Column N=0 of C/D tile: lane 0 holds M=0..7, lane 16 holds M=8..15
    if (mrow == 0) {
        #pragma unroll
        for (int i = 0; i < 8; ++i) red[wave][half * 8 + i] = acc[i];
    }
    __syncthreads();
    if (tid < 16) {
        float sum = 0.0f;
        #pragma unroll
        for (int w = 0; w < 8; ++w) sum += red[w][tid];
        const int r = m0 + tid;
        gates[r] = sum + b_ih[r] + b_hh[r];
    }
}

// ---------------------------------------------------------------------------
// Kernel 2: LSTM pointwise: c = sig(f)*c0 + sig(i)*tanh(g); h = sig(o)*tanh(c)
// ---------------------------------------------------------------------------
__global__ __launch_bounds__(256)
void lstm_pointwise(const float* __restrict__ gates, const float* __restrict__ c0,
                    float* __restrict__ h)
{
    const int j = blockIdx.x * blockDim.x + threadIdx.x;
    if (j >= Hd) return;
    const float gi = gates[0 * Hd + j];
    const float gf = gates[1 * Hd + j];
    const float gg = gates[2 * Hd + j];
    const float go = gates[3 * Hd + j];
    const float i = 1.0f / (1.0f + __expf(-gi));
    const float f = 1.0f / (1.0f + __expf(-gf));
    const float g = tanhf(gg);
    const float o = 1.0f / (1.0f + __expf(-go));
    const float c = f * c0[j] + i * g;
    h[j] = o * tanhf(c);
}

// ---------------------------------------------------------------------------
// Kernel 3: MLP head: out = sigmoid(W2 @ relu(W1 @ h + b1) + b2)  (130 outputs)
// ---------------------------------------------------------------------------
__global__ __launch_bounds__(256)
void mlp_head(const float* __restrict__ h,  const float* __restrict__ W1,
              const float* __restrict__ b1, const float* __restrict__ W2,
              const float* __restrict__ b2, float* __restrict__ out)
{
    __shared__ float part[32][8];
    __shared__ float z[32];
    const int tid = threadIdx.x;
    const int row = tid >> 3;
    const int seg = tid & 7;

    float s = 0.0f;
    const float* wr = W1 + (long long)row * Hd + seg * 512;
    const float* hv = h + seg * 512;
    #pragma unroll 4
    for (int j = 0; j < 512; j += 4) {
        s += wr[j + 0] * hv[j + 0] + wr[j + 1] * hv[j + 1]
           + wr[j + 2] * hv[j + 2] + wr[j + 3] * hv[j + 3];
    }
    part[row][seg] = s;
    __syncthreads();

    if (tid < 32) {
        float tz = b1[tid];
        #pragma unroll
        for (int k = 0; k < 8; ++k) tz += part[tid][k];
        z[tid] = tz > 0.0f ? tz : 0.0f;
    }
    __syncthreads();

    if (tid < 130) {                        // OUT = 2 + 128
        float tz = b2[tid];
        const float* w2 = W2 + tid * 32;
        #pragma unroll
        for (int k = 0; k < 32; ++k) tz += w2[k] * z[k];
        out[tid] = 1.0f / (1.0f + __expf(-tz));
    }
}

// ---------------------------------------------------------------------------
extern "C" void kernel_launch(void* const* d_in, const int* in_sizes, int n_in,
                              void* d_out, int out_size, void* d_ws, size_t ws_size,
                              hipStream_t stream)
{
    const float* x    = (const float*)d_in[0];
    const float* h0   = (const float*)d_in[1];
    const float* c0   = (const float*)d_in[2];
    const float* W_ih = (const float*)d_in[3];
    const float* W_hh = (const float*)d_in[4];
    const float* b_ih = (const float*)d_in[5];
    const float* b_hh = (const float*)d_in[6];
    const float* W1   = (const float*)d_in[7];
    const float* b1   = (const float*)d_in[8];
    const float* W2   = (const float*)d_in[9];
    const float* b2   = (const float*)d_in[10];
    float* out = (float*)d_out;

    float* gates = (float*)d_ws;            // 16384 floats
    float* h     = gates + RTOT;            // 4096 floats

    lstm_gates_wmma<<<RTOT / 16, 256, 0, stream>>>(x, h0, W_ih, W_hh, b_ih, b_hh, gates);
    lstm_pointwise<<<Hd / 256, 256, 0, stream>>>(gates, c0, h);
    mlp_head<<<1, 256, 0, stream>>>(h, W1, b1, W2, b2, out);
}